// StaticGNN_7679401525531
// MI455X (gfx1250) — compile-verified
//
#include <hip/hip_runtime.h>
#include <cmath>

typedef __attribute__((ext_vector_type(2))) float v2f;
typedef __attribute__((ext_vector_type(8))) float v8f;

#define GNN_N   100000
#define GNN_E   1600000
#define GNN_IN  128
#define GNN_HID 64
#define GNN_MID 32
#define GNN_EPS 1e-5f

// ---------------------------------------------------------------------------
// Utility: fill buffer with a constant
// ---------------------------------------------------------------------------
__global__ void fill_kernel(float* __restrict__ p, int n, float v) {
    int i = blockIdx.x * blockDim.x + threadIdx.x;
    if (i < n) p[i] = v;
}

// ---------------------------------------------------------------------------
// Degree accumulation: deg[dst[e]] += 1
// ---------------------------------------------------------------------------
__global__ void deg_kernel(const int* __restrict__ dst, float* __restrict__ deg, int e) {
    int i = blockIdx.x * blockDim.x + threadIdx.x;
    if (i < e) atomicAdd(&deg[dst[i]], 1.0f);
}

// dinv[i] = rsqrt(deg[i] + 1)   (self-loop adds 1, so deg >= 1)
__global__ void dinv_kernel(float* __restrict__ d, int n) {
    int i = blockIdx.x * blockDim.x + threadIdx.x;
    if (i < n) d[i] = rsqrtf(d[i] + 1.0f);
}

// ---------------------------------------------------------------------------
// Dense GEMM  C[n x F] = A[n x K] * W[K x F]  via V_WMMA_F32_16X16X4_F32
// block = 256 threads = 8 waves; each wave owns one 16x16 C tile.
// F must be a multiple of 16 with (F/16) dividing 8; n a multiple of 16.
// Fragment layouts per CDNA5 ISA 7.12.2:
//   A 16x4 f32 : lanes 0-15 -> {K=0,K=1}, lanes 16-31 -> {K=2,K=3}, M = lane&15
//   B 4x16 f32 : mirrored, N = lane&15
//   C/D 16x16  : lane<16 -> M=v, N=lane ; lane>=16 -> M=v+8, N=lane-16
// ---------------------------------------------------------------------------
__global__ void gemm_wmma_kernel(const float* __restrict__ A,
                                 const float* __restrict__ W,
                                 float* __restrict__ C,
                                 int n, int K, int F) {
    const int wave = threadIdx.x >> 5;
    const int lane = threadIdx.x & 31;
    const int colTiles = F >> 4;                 // 4 for F=64
    const int rowTilesPerBlock = 8 / colTiles;   // 2 for F=64
    const int rowTile = blockIdx.x * rowTilesPerBlock + (wave / colTiles);
    const int colTile = wave % colTiles;
    const int row0 = rowTile << 4;
    const int col0 = colTile << 4;
    if (row0 >= n) return;                       // uniform within a wave

    const int half = lane >> 4;                  // 0: K pair {0,1}, 1: {2,3}
    const int lr   = lane & 15;

    v8f acc = {};
    const float* Arow = A + (size_t)(row0 + lr) * K;
    const float* Wcol = W + (col0 + lr);

    for (int k = 0; k < K; k += 4) {
        const int k0 = k + 2 * half;
        v2f a, b;
        a.x = Arow[k0];
        a.y = Arow[k0 + 1];
        b.x = Wcol[(size_t)k0 * F];
        b.y = Wcol[(size_t)(k0 + 1) * F];
        acc = __builtin_amdgcn_wmma_f32_16x16x4_f32(
            /*neg_a=*/false, a, /*neg_b=*/false, b,
            /*c_mod=*/(short)0, acc, /*reuse_a=*/false, /*reuse_b=*/false);
    }

    float* Crow = C + (size_t)row0 * F + col0 + lr;
#pragma unroll
    for (int v = 0; v < 8; ++v) {
        Crow[(size_t)(v + 8 * half) * F] = acc[v];
    }
}

// ---------------------------------------------------------------------------
// Edge aggregation: agg[dst] += dinv[src]*dinv[dst] * h[src]
// 16 lanes per edge, each lane handles a float4 of the 64 features.
// ---------------------------------------------------------------------------
__global__ void agg_kernel(const int* __restrict__ src, const int* __restrict__ dst,
                           const float* __restrict__ dinv,
                           const float* __restrict__ h, float* __restrict__ agg, int e) {
    int idx = blockIdx.x * blockDim.x + threadIdx.x;
    if (idx >= e * 16) return;
    const int ed = idx >> 4;
    const int q  = idx & 15;
    const int s = src[ed];
    const int d = dst[ed];
    const float nrm = dinv[s] * dinv[d];
    const float4 v = ((const float4*)(h + (size_t)s * GNN_HID))[q];
    float* o = agg + (size_t)d * GNN_HID + q * 4;
    atomicAdd(o + 0, nrm * v.x);
    atomicAdd(o + 1, nrm * v.y);
    atomicAdd(o + 2, nrm * v.z);
    atomicAdd(o + 3, nrm * v.w);
}

// ---------------------------------------------------------------------------
// Epilogue: out = relu(BN(agg + dinv^2 * h + bias))
// One thread per (node, float4-quad); 16 quads per node.
// ---------------------------------------------------------------------------
__global__ void epilogue_kernel(const float* __restrict__ agg, const float* __restrict__ h,
                                const float* __restrict__ dinv,
                                const float* __restrict__ b, const float* __restrict__ g,
                                const float* __restrict__ be, const float* __restrict__ m,
                                const float* __restrict__ v,
                                float* __restrict__ out, int n) {
    int idx = blockIdx.x * blockDim.x + threadIdx.x;
    if (idx >= n * 16) return;
    const int i = idx >> 4;
    const int q = idx & 15;
    const float di2 = dinv[i] * dinv[i];
    const float4 av = ((const float4*)(agg + (size_t)i * GNN_HID))[q];
    const float4 hv = ((const float4*)(h + (size_t)i * GNN_HID))[q];
    const float4 bv  = ((const float4*)b)[q];
    const float4 gv  = ((const float4*)g)[q];
    const float4 bev = ((const float4*)be)[q];
    const float4 mv  = ((const float4*)m)[q];
    const float4 vv  = ((const float4*)v)[q];
    float4 r;
    r.x = fmaxf((av.x + di2 * hv.x + bv.x - mv.x) * rsqrtf(vv.x + GNN_EPS) * gv.x + bev.x, 0.0f);
    r.y = fmaxf((av.y + di2 * hv.y + bv.y - mv.y) * rsqrtf(vv.y + GNN_EPS) * gv.y + bev.y, 0.0f);
    r.z = fmaxf((av.z + di2 * hv.z + bv.z - mv.z) * rsqrtf(vv.z + GNN_EPS) * gv.z + bev.z, 0.0f);
    r.w = fmaxf((av.w + di2 * hv.w + bv.w - mv.w) * rsqrtf(vv.w + GNN_EPS) * gv.w + bev.w, 0.0f);
    ((float4*)(out + (size_t)i * GNN_HID))[q] = r;
}

// ---------------------------------------------------------------------------
// Classifier head: out[i] = sigmoid(relu(h[i] @ Wc1 + bc1) @ Wc2 + bc2)
// Wc1 (64x32) cached in LDS; every lane reads the same LDS address -> broadcast.
// ---------------------------------------------------------------------------
__global__ void classifier_kernel(const float* __restrict__ h,
                                  const float* __restrict__ Wc1, const float* __restrict__ bc1,
                                  const float* __restrict__ Wc2, const float* __restrict__ bc2,
                                  float* __restrict__ out, int n) {
    __shared__ float sW1[GNN_HID * GNN_MID];
    __shared__ float sW2[GNN_MID];
    __shared__ float sb1[GNN_MID];
    for (int t = threadIdx.x; t < GNN_HID * GNN_MID; t += blockDim.x) sW1[t] = Wc1[t];
    if (threadIdx.x < GNN_MID) {
        sW2[threadIdx.x] = Wc2[threadIdx.x];
        sb1[threadIdx.x] = bc1[threadIdx.x];
    }
    __syncthreads();

    int i = blockIdx.x * blockDim.x + threadIdx.x;
    if (i >= n) return;

    float x[GNN_HID];
    const float4* hp = (const float4*)(h + (size_t)i * GNN_HID);
#pragma unroll
    for (int q = 0; q < 16; ++q) {
        float4 t = hp[q];
        x[4 * q + 0] = t.x; x[4 * q + 1] = t.y;
        x[4 * q + 2] = t.z; x[4 * q + 3] = t.w;
    }
    float z = bc2[0];
    for (int mm = 0; mm < GNN_MID; ++mm) {
        float s = sb1[mm];
#pragma unroll
        for (int hh = 0; hh < GNN_HID; ++hh)
            s = fmaf(x[hh], sW1[hh * GNN_MID + mm], s);
        z = fmaf(fmaxf(s, 0.0f), sW2[mm], z);
    }
    out[i] = 1.0f / (1.0f + __expf(-z));
}

// ---------------------------------------------------------------------------
// Launcher
// Workspace layout (floats): dinv[N] | bufA[N*64] | bufB[N*64] | bufC[N*64]
// total = 4*N*(1+192) bytes ~= 77.2 MB
// ---------------------------------------------------------------------------
extern "C" void kernel_launch(void* const* d_in, const int* in_sizes, int n_in,
                              void* d_out, int out_size, void* d_ws, size_t ws_size,
                              hipStream_t stream) {
    const int N = GNN_N, E = GNN_E;

    const float* x   = (const float*)d_in[0];
    const int*   ei  = (const int*)d_in[1];
    const float* W1  = (const float*)d_in[2];
    const float* b1  = (const float*)d_in[3];
    const float* g1  = (const float*)d_in[4];
    const float* be1 = (const float*)d_in[5];
    const float* m1  = (const float*)d_in[6];
    const float* v1  = (const float*)d_in[7];
    const float* W2  = (const float*)d_in[8];
    const float* b2  = (const float*)d_in[9];
    const float* g2  = (const float*)d_in[10];
    const float* be2 = (const float*)d_in[11];
    const float* m2  = (const float*)d_in[12];
    const float* v2  = (const float*)d_in[13];
    const float* Wc1 = (const float*)d_in[14];
    const float* bc1 = (const float*)d_in[15];
    const float* Wc2 = (const float*)d_in[16];
    const float* bc2 = (const float*)d_in[17];
    float* out = (float*)d_out;

    const int* src = ei;
    const int* dst = ei + E;

    float* dinv = (float*)d_ws;
    float* bufA = dinv + N;
    float* bufB = bufA + (size_t)N * GNN_HID;
    float* bufC = bufB + (size_t)N * GNN_HID;

    const int TB = 256;
    const int gN    = (N + TB - 1) / TB;
    const int gE    = (E + TB - 1) / TB;
    const int gNH   = (N * 16 + TB - 1) / TB;        // node*featquad
    const int gEH   = (E * 16 + TB - 1) / TB;        // edge*featquad
    const int gNF   = (N * GNN_HID + TB - 1) / TB;
    const int gGemm = (N + 31) / 32;                 // 2 row-tiles (32 rows) per block

    // ---- degrees & dinv ----
    fill_kernel<<<gN, TB, 0, stream>>>(dinv, N, 0.0f);
    deg_kernel<<<gE, TB, 0, stream>>>(dst, dinv, E);
    dinv_kernel<<<gN, TB, 0, stream>>>(dinv, N);

    // ---- layer 1: bufA = x @ W1 ; bufB = scatter ; bufC = relu(bn(...)) ----
    gemm_wmma_kernel<<<gGemm, TB, 0, stream>>>(x, W1, bufA, N, GNN_IN, GNN_HID);
    fill_kernel<<<gNF, TB, 0, stream>>>(bufB, N * GNN_HID, 0.0f);
    agg_kernel<<<gEH, TB, 0, stream>>>(src, dst, dinv, bufA, bufB, E);
    epilogue_kernel<<<gNH, TB, 0, stream>>>(bufB, bufA, dinv, b1, g1, be1, m1, v1, bufC, N);

    // ---- layer 2: bufA = bufC @ W2 ; bufB = scatter ; bufC = relu(bn(...)) ----
    gemm_wmma_kernel<<<gGemm, TB, 0, stream>>>(bufC, W2, bufA, N, GNN_HID, GNN_HID);
    fill_kernel<<<gNF, TB, 0, stream>>>(bufB, N * GNN_HID, 0.0f);
    agg_kernel<<<gEH, TB, 0, stream>>>(src, dst, dinv, bufA, bufB, E);
    epilogue_kernel<<<gNH, TB, 0, stream>>>(bufB, bufA, dinv, b2, g2, be2, m2, v2, bufC, N);

    // ---- classifier head ----
    classifier_kernel<<<gN, TB, 0, stream>>>(bufC, Wc1, bc1, Wc2, bc2, out, N);
}